// RoIPooling_26414048871046
// MI455X (gfx1250) — compile-verified
//
#include <hip/hip_runtime.h>

// crop_and_resize (RoIAlign, bilinear, 7x7 grid) for feats (8,100,136,256) f32.
// One wave32 = 16 sample points. Bilinear 4-tap blend runs on the matrix unit
// as 16 accumulated rank-4 updates of V_WMMA_F32_16X16X4_F32:
//   D[m, n] = sum_p A_p[m,k] * B_p[k,n],  A_p row p = [w_tl,w_tr,w_bl,w_br]
// giving D[point m][channel n] exactly in f32. Channel blocks are software-
// pipelined (ping-pong buffers + sched_barrier) so the WMMA chain of block cb
// overlaps the 32 in-flight gathers of block cb+1.

typedef __attribute__((ext_vector_type(2))) float v2f;
typedef __attribute__((ext_vector_type(8))) float v8f;

#define H 100
#define W 136
#define C 256
#define CROP 7
#define PER_BOX (CROP * CROP)        // 49
#define IMG_ELEMS (H * W * C)        // 3,481,600

// Issue the 32 coalesced b32 gathers for channel block CB (immediate offsets
// after full unroll: offset = CB*64 bytes).
#define LOAD_BLK(buf0, buf1, CB)                                               \
  _Pragma("unroll")                                                            \
  for (int p = 0; p < 16; ++p) {                                               \
    buf0[p] = feats[vo0[p] + (CB) * 16];                                       \
    buf1[p] = feats[vo1[p] + (CB) * 16];                                       \
  }

// 16 accumulated rank-4 WMMAs for channel block CB, then 8 b32 stores.
#define WMMA_STORE(buf0, buf1, CB)                                             \
  {                                                                            \
    v8f acc = {};                                                              \
    _Pragma("unroll")                                                          \
    for (int p = 0; p < 16; ++p) {                                             \
      v2f a, b;                                                                \
      a.x = (m == p) ? wA0 : 0.0f;                                             \
      a.y = (m == p) ? wA1 : 0.0f;                                             \
      b.x = buf0[p];                                                           \
      b.y = buf1[p];                                                           \
      acc = __builtin_amdgcn_wmma_f32_16x16x4_f32(                             \
          /*neg_a=*/false, a, /*neg_b=*/false, b,                              \
          /*c_mod=*/(short)0, acc, /*reuse_a=*/false, /*reuse_b=*/false);      \
    }                                                                          \
    _Pragma("unroll")                                                          \
    for (int j = 0; j < 8; ++j) outw[od + j * C + (CB) * 16] = acc[j];         \
  }

__global__ __launch_bounds__(256) void crop_resize_wmma_kernel(
    const float* __restrict__ feats,
    const float* __restrict__ boxes,
    const int* __restrict__ box_ind,
    float* __restrict__ out,
    int total_pts) {
  const int lane = threadIdx.x & 31;
  const int m = lane & 15;           // point-in-wave / channel-in-block id
  const bool lo = lane < 16;

  const int wave = (blockIdx.x * blockDim.x + threadIdx.x) >> 5;
  const int pbase = wave * 16;
  if (pbase >= total_pts) return;    // wave-uniform: EXEC stays all-1s below

  // ---- per-point metadata (lanes m and m+16 both compute point pbase+m) ----
  int pt = pbase + m;
  const bool pt_ok = pt < total_pts; // always true: 16 | 2000*49
  pt = pt_ok ? pt : (total_pts - 1);

  const int box = pt / PER_BOX;
  const int r = pt - box * PER_BOX;
  const int iy = r / CROP;
  const int ix = r - iy * CROP;

  const float4 bx = ((const float4*)boxes)[box];   // y1, x1, y2, x2
  const int bi = box_ind[box];

  const float fy = (float)(H - 1);   // 99
  const float fx = (float)(W - 1);   // 135
  const float ys = bx.x * fy + (float)iy * ((bx.z - bx.x) * (fy / (float)(CROP - 1)));
  const float xs = bx.y * fx + (float)ix * ((bx.w - bx.y) * (fx / (float)(CROP - 1)));

  const bool valid = pt_ok & (ys >= 0.0f) & (ys <= fy) & (xs >= 0.0f) & (xs <= fx);
  const float ysc = fminf(fmaxf(ys, 0.0f), fy);
  const float xsc = fminf(fmaxf(xs, 0.0f), fx);
  const int y0 = (int)floorf(ysc);
  const int x0 = (int)floorf(xsc);
  const int y1i = min(y0 + 1, H - 1);
  const int x1i = min(x0 + 1, W - 1);
  const float ly = ysc - (float)y0;
  const float lx = xsc - (float)x0;

  const float w_tl = valid ? (1.0f - ly) * (1.0f - lx) : 0.0f;
  const float w_tr = valid ? (1.0f - ly) * lx          : 0.0f;
  const float w_bl = valid ? ly * (1.0f - lx)          : 0.0f;
  const float w_br = valid ? ly * lx                   : 0.0f;

  const int ibase = bi * IMG_ELEMS;
  const int o_tl = ibase + (y0  * W + x0 ) * C;      // element offsets
  const int o_tr = ibase + (y0  * W + x1i) * C;
  const int o_bl = ibase + (y1i * W + x0 ) * C;
  const int o_br = ibase + (y1i * W + x1i) * C;

  // A-matrix lane layout (16x4 f32): lanes 0-15 hold K=0,1 ; lanes 16-31 K=2,3.
  const float wA0 = lo ? w_tl : w_bl;  // K=0 (lo) / K=2 (hi)
  const float wA1 = lo ? w_tr : w_br;  // K=1 (lo) / K=3 (hi)

  // ---- broadcast each point's corner-row offsets; bake in lane's channel ----
  // B-matrix 4x16: VGPR0 = rows K=0 (lanes 0-15) / K=2 (lanes 16-31),
  //                VGPR1 = rows K=1 / K=3.  lo lanes gather tl/tr, hi lanes
  //                gather bl/br, at channel n = cb*16 + m.
  int vo0[16], vo1[16];
#pragma unroll
  for (int p = 0; p < 16; ++p) {
    const int s_tl = __builtin_amdgcn_readlane(o_tl, p);
    const int s_tr = __builtin_amdgcn_readlane(o_tr, p);
    const int s_bl = __builtin_amdgcn_readlane(o_bl, p);
    const int s_br = __builtin_amdgcn_readlane(o_br, p);
    vo0[p] = (lo ? s_tl : s_bl) + m;
    vo1[p] = (lo ? s_tr : s_br) + m;
  }

  // D layout (16x16 f32): VGPR j -> point j (lanes 0-15) / point j+8 (lanes 16-31)
  float* outw = out + (long)pbase * C;
  const int od = (lo ? 0 : 8 * C) + m;

  if (pbase + 16 <= total_pts) {
    // Fast path (always taken at these shapes): unguarded stores, fully
    // unrolled, ping-pong double-buffered channel blocks.
    float bA0[16], bA1[16], bB0[16], bB1[16];
    LOAD_BLK(bA0, bA1, 0);
#pragma unroll
    for (int cb = 0; cb < C / 16; cb += 2) {
      LOAD_BLK(bB0, bB1, cb + 1);              // prefetch odd block
      __builtin_amdgcn_sched_barrier(0);
      WMMA_STORE(bA0, bA1, cb);                // consume even block
      __builtin_amdgcn_sched_barrier(0);
      if (cb + 2 < C / 16) {
        LOAD_BLK(bA0, bA1, cb + 2);            // prefetch next even block
      }
      __builtin_amdgcn_sched_barrier(0);
      WMMA_STORE(bB0, bB1, cb + 1);            // consume odd block
      __builtin_amdgcn_sched_barrier(0);
    }
  } else {
    // Tail fallback (unreachable for 2000 boxes): per-lane guarded stores.
    for (int cb = 0; cb < C / 16; ++cb) {
      float b0v[16], b1v[16];
      LOAD_BLK(b0v, b1v, cb);
      v8f acc = {};
#pragma unroll
      for (int p = 0; p < 16; ++p) {
        v2f a, b;
        a.x = (m == p) ? wA0 : 0.0f;
        a.y = (m == p) ? wA1 : 0.0f;
        b.x = b0v[p];
        b.y = b1v[p];
        acc = __builtin_amdgcn_wmma_f32_16x16x4_f32(
            false, a, false, b, (short)0, acc, false, false);
      }
#pragma unroll
      for (int j = 0; j < 8; ++j) {
        const int pj = pbase + (lo ? j : j + 8);
        if (pj < total_pts) outw[od + j * C + cb * 16] = acc[j];
      }
    }
  }
}

extern "C" void kernel_launch(void* const* d_in, const int* in_sizes, int n_in,
                              void* d_out, int out_size, void* d_ws, size_t ws_size,
                              hipStream_t stream) {
  const float* feats = (const float*)d_in[0];
  const float* boxes = (const float*)d_in[1];
  const int* box_ind = (const int*)d_in[2];
  float* out = (float*)d_out;

  const int n_boxes = in_sizes[2];              // 2000
  const int total_pts = n_boxes * PER_BOX;      // 98,000
  const int waves = (total_pts + 15) / 16;      // 6,125
  const int threads = waves * 32;
  const int block = 256;
  const int grid = (threads + block - 1) / block;

  crop_resize_wmma_kernel<<<grid, block, 0, stream>>>(feats, boxes, box_ind, out,
                                                      total_pts);
}